// GIFT_12910671691875
// MI455X (gfx1250) — compile-verified
//
#include <hip/hip_runtime.h>
#include <math.h>

typedef __attribute__((ext_vector_type(2))) float v2f;
typedef __attribute__((ext_vector_type(8))) float v8f;

#define O_DIM 128

__device__ __forceinline__ void atomAddF(float* p, float v) {
    __hip_atomic_fetch_add(p, v, __ATOMIC_RELAXED, __HIP_MEMORY_SCOPE_AGENT);
}

// ---------------------------------------------------------------------------
// GEMM: out[N,128] = relu(X[N,K] @ W[K,128] + b)   (K in {128,256}, mult of 32)
// Block: 256 threads = 8 waves. Block tile: 128 rows x 128 cols.
// Wave w computes rows [m0+16w, m0+16w+16) x all 128 cols via 8 WMMA tiles.
// WMMA: V_WMMA_F32_16X16X4_F32, fp32 in/out (matches reference numerics).
// LDS layouts are chosen so both A and B fragments are single conflict-free
// ds_load_b64 ops (K-pairs contiguous), eliminating register-pairing movs.
// ---------------------------------------------------------------------------
__global__ __launch_bounds__(256)
void gemm_bias_relu(const float* __restrict__ X, const float* __restrict__ W,
                    const float* __restrict__ bias, float* __restrict__ out,
                    int N, int K) {
    __shared__ float  As[128][36];    // 128x32 chunk of X; stride 36 floats
    __shared__ float2 Bs2[16][144];   // K-pair p -> {W[2p][n], W[2p+1][n]}; stride 144 f2

    const int tid = threadIdx.x;
    const int wv  = tid >> 5;
    const int ln  = tid & 31;
    const int m0  = blockIdx.x * 128;

    v8f acc[8];
#pragma unroll
    for (int t = 0; t < 8; ++t)
#pragma unroll
        for (int j = 0; j < 8; ++j) acc[t][j] = 0.0f;

    const int mrow = wv * 16 + (ln & 15);
    const int lnn  = ln & 15;
    const int khalf = (ln >> 4) << 1;      // lanes 16-31 take K=+2,+3

    for (int k0 = 0; k0 < K; k0 += 32) {
        // ---- cooperative stage: X[128x32] -> As (float4 global, float4 LDS) ----
#pragma unroll
        for (int i = 0; i < 4; ++i) {
            int f   = tid + i * 256;          // 0..1023 float4 slots
            int row = f >> 3;                 // 8 float4 per row
            int c4  = f & 7;
            float4 v = make_float4(0.f, 0.f, 0.f, 0.f);
            if (m0 + row < N)
                v = *(const float4*)(X + (size_t)(m0 + row) * K + k0 + c4 * 4);
            *(float4*)&As[row][c4 * 4] = v;   // 36*4=144B stride, 16B aligned
        }
        // ---- cooperative stage: W[32x128] -> Bs2 with K-pair interleave ----
#pragma unroll
        for (int i = 0; i < 4; ++i) {
            int f   = tid + i * 256;
            int row = f >> 5;                 // K index within chunk, 0..31
            int c4  = f & 31;                 // float4 col group
            float4 v = *(const float4*)(W + (size_t)(k0 + row) * O_DIM + c4 * 4);
            float* bp = (float*)&Bs2[row >> 1][c4 * 4] + (row & 1);
            bp[0] = v.x;                      // component .x/.y by K parity
            bp[2] = v.y;
            bp[4] = v.z;
            bp[6] = v.w;
        }
        __syncthreads();

        // ---- 8 k-steps of V_WMMA_F32_16X16X4_F32 per chunk ----
#pragma unroll
        for (int kk = 0; kk < 32; kk += 4) {
            const int kk2 = kk + khalf;
            const int p   = kk2 >> 1;
            v2f a = *(const v2f*)&As[mrow][kk2];          // one ds_load_b64
#pragma unroll
            for (int t = 0; t < 8; ++t) {
                v2f b = *(const v2f*)&Bs2[p][t * 16 + lnn]; // one ds_load_b64
                acc[t] = __builtin_amdgcn_wmma_f32_16x16x4_f32(
                    false, a, false, b, (short)0, acc[t], false, false);
            }
        }
        __syncthreads();
    }

    // ---- epilogue: bias + relu + store (C layout: vgpr i -> rows i / i+8) ----
    const int mtile = m0 + wv * 16;
    if (mtile < N) {
        const int rowoff = (ln >> 4) << 3;
#pragma unroll
        for (int t = 0; t < 8; ++t) {
            const int col = t * 16 + lnn;
            const float bv = bias[col];
#pragma unroll
            for (int i = 0; i < 8; ++i) {
                float v = acc[t][i] + bv;
                out[(size_t)(mtile + rowoff + i) * O_DIM + col] = v > 0.f ? v : 0.f;
            }
        }
    }
}

// ---------------------------------------------------------------------------
// SpMM scatter: out[dst[e], :] += w[e] * x[src[e], :], D=128.
// One wave per edge; lane handles a float4 of features.
// ---------------------------------------------------------------------------
__global__ __launch_bounds__(256)
void spmm128(const int* __restrict__ src, const int* __restrict__ dst,
             const float* __restrict__ w, const float* __restrict__ x,
             float* __restrict__ out, int E) {
    int e = blockIdx.x * 8 + (threadIdx.x >> 5);
    if (e >= E) return;
    const int ln = threadIdx.x & 31;
    const int s = src[e];
    const int d = dst[e];
    const float wt = w[e];
    float4 v = ((const float4*)(x + (size_t)s * 128))[ln];
    float* o = out + (size_t)d * 128 + ln * 4;
    atomAddF(o + 0, wt * v.x);
    atomAddF(o + 1, wt * v.y);
    atomAddF(o + 2, wt * v.z);
    atomAddF(o + 3, wt * v.w);
}

// ---------------------------------------------------------------------------
// SpMM scatter with implicit concat: out row = [xa(128) | xb(128)], D=256.
// ---------------------------------------------------------------------------
__global__ __launch_bounds__(256)
void spmm_concat256(const int* __restrict__ src, const int* __restrict__ dst,
                    const float* __restrict__ w, const float* __restrict__ xa,
                    const float* __restrict__ xb, float* __restrict__ out, int E) {
    int e = blockIdx.x * 8 + (threadIdx.x >> 5);
    if (e >= E) return;
    const int ln = threadIdx.x & 31;
    const int s = src[e];
    const int d = dst[e];
    const float wt = w[e];
    float4 va = ((const float4*)(xa + (size_t)s * 128))[ln];
    float4 vb = ((const float4*)(xb + (size_t)s * 128))[ln];
    float* oa = out + (size_t)d * 256 + ln * 4;
    float* ob = oa + 128;
    atomAddF(oa + 0, wt * va.x);
    atomAddF(oa + 1, wt * va.y);
    atomAddF(oa + 2, wt * va.z);
    atomAddF(oa + 3, wt * va.w);
    atomAddF(ob + 0, wt * vb.x);
    atomAddF(ob + 1, wt * vb.y);
    atomAddF(ob + 2, wt * vb.z);
    atomAddF(ob + 3, wt * vb.w);
}

// ---------------------------------------------------------------------------
// out[row] = [ r0[row]/(||r0||+eps) (128) | r1[row]/(||r1||+eps) (256) ]
// One wave per row; wave32 __shfl_xor reduction.
// ---------------------------------------------------------------------------
__global__ __launch_bounds__(256)
void l2norm_concat(const float* __restrict__ r0, const float* __restrict__ r1,
                   float* __restrict__ out, int N) {
    int row = blockIdx.x * 8 + (threadIdx.x >> 5);
    if (row >= N) return;
    const int ln = threadIdx.x & 31;

    float4 v0  = ((const float4*)(r0 + (size_t)row * 128))[ln];
    float4 v1a = ((const float4*)(r1 + (size_t)row * 256))[ln];
    float4 v1b = ((const float4*)(r1 + (size_t)row * 256))[ln + 32];

    float s0 = v0.x * v0.x + v0.y * v0.y + v0.z * v0.z + v0.w * v0.w;
    float s1 = v1a.x * v1a.x + v1a.y * v1a.y + v1a.z * v1a.z + v1a.w * v1a.w
             + v1b.x * v1b.x + v1b.y * v1b.y + v1b.z * v1b.z + v1b.w * v1b.w;
#pragma unroll
    for (int off = 16; off > 0; off >>= 1) {
        s0 += __shfl_xor(s0, off, 32);
        s1 += __shfl_xor(s1, off, 32);
    }
    const float i0 = 1.0f / (sqrtf(s0) + 1e-9f);
    const float i1 = 1.0f / (sqrtf(s1) + 1e-9f);

    float* o = out + (size_t)row * 384;
    float4 w0 = make_float4(v0.x * i0, v0.y * i0, v0.z * i0, v0.w * i0);
    float4 wa = make_float4(v1a.x * i1, v1a.y * i1, v1a.z * i1, v1a.w * i1);
    float4 wb = make_float4(v1b.x * i1, v1b.y * i1, v1b.z * i1, v1b.w * i1);
    *(float4*)(o + ln * 4)        = w0;
    *(float4*)(o + 128 + ln * 4)  = wa;
    *(float4*)(o + 256 + ln * 4)  = wb;
}

// ---------------------------------------------------------------------------
extern "C" void kernel_launch(void* const* d_in, const int* in_sizes, int n_in,
                              void* d_out, int out_size, void* d_ws, size_t ws_size,
                              hipStream_t stream) {
    const float* x1       = (const float*)d_in[0];
    const float* x2       = (const float*)d_in[1];
    const float* word_emb = (const float*)d_in[2];
    const float* W1a = (const float*)d_in[3];
    const float* b1a = (const float*)d_in[4];
    const float* W1b = (const float*)d_in[5];
    const float* b1b = (const float*)d_in[6];
    const float* W2a = (const float*)d_in[7];
    const float* b2a = (const float*)d_in[8];
    const float* W2b = (const float*)d_in[9];
    const float* b2b = (const float*)d_in[10];
    const int*   e11_src = (const int*)d_in[11];
    const int*   e11_dst = (const int*)d_in[12];
    const float* e11_w   = (const float*)d_in[13];
    const int*   e22_src = (const int*)d_in[14];
    const int*   e22_dst = (const int*)d_in[15];
    const float* e22_w   = (const float*)d_in[16];
    const int*   e01_src = (const int*)d_in[17];
    const int*   e01_dst = (const int*)d_in[18];
    const float* e01_w   = (const float*)d_in[19];
    const int*   e02_src = (const int*)d_in[20];
    const int*   e02_dst = (const int*)d_in[21];
    const float* e02_w   = (const float*)d_in[22];

    const int N0 = 10000, N1 = 50000, N2 = 30000;
    const int E11 = in_sizes[11], E22 = in_sizes[14];
    const int E01 = in_sizes[17], E02 = in_sizes[20];

    // ---- workspace layout (floats) ----
    float* ws = (float*)d_ws;
    size_t off = 0;
    float* l1_1 = ws + off; off += (size_t)N1 * 128;
    float* h11  = ws + off; off += (size_t)N1 * 128;
    float* l2_1 = ws + off; off += (size_t)N1 * 128;
    float* l1_2 = ws + off; off += (size_t)N2 * 128;
    float* h22  = ws + off; off += (size_t)N2 * 128;
    float* l2_2 = ws + off; off += (size_t)N2 * 128;
    float* r0   = ws + off; off += (size_t)N0 * 128;
    float* r1   = ws + off; off += (size_t)N0 * 256;
    float* r0s  = ws + off; off += (size_t)N0 * 128;
    float* r1s  = ws + off; off += (size_t)N0 * 256;

    float* doc     = (float*)d_out;
    float* doc_svd = doc + (size_t)N0 * 384;

    const dim3 blk(256);
    auto gemm_grid = [](int n) { return dim3((n + 127) / 128); };
    auto edge_grid = [](int e) { return dim3((e + 7) / 8); };

    // type-1 tower
    gemm_bias_relu<<<gemm_grid(N1), blk, 0, stream>>>(x1, W1a, b1a, l1_1, N1, 256);
    hipMemsetAsync(h11, 0, (size_t)N1 * 128 * sizeof(float), stream);
    spmm128<<<edge_grid(E11), blk, 0, stream>>>(e11_src, e11_dst, e11_w, l1_1, h11, E11);
    gemm_bias_relu<<<gemm_grid(N1), blk, 0, stream>>>(h11, W1b, b1b, l2_1, N1, 128);

    // type-2 tower
    gemm_bias_relu<<<gemm_grid(N2), blk, 0, stream>>>(x2, W2a, b2a, l1_2, N2, 256);
    hipMemsetAsync(h22, 0, (size_t)N2 * 128 * sizeof(float), stream);
    spmm128<<<edge_grid(E22), blk, 0, stream>>>(e22_src, e22_dst, e22_w, l1_2, h22, E22);
    gemm_bias_relu<<<gemm_grid(N2), blk, 0, stream>>>(h22, W2b, b2b, l2_2, N2, 128);

    // aggregate to documents (r0,r1,r0s,r1s are contiguous -> one memset)
    hipMemsetAsync(r0, 0, (size_t)N0 * (128 + 256 + 128 + 256) * sizeof(float), stream);
    spmm128<<<edge_grid(E01), blk, 0, stream>>>(e01_src, e01_dst, e01_w, l2_1, r0, E01);
    spmm_concat256<<<edge_grid(E02), blk, 0, stream>>>(e02_src, e02_dst, e02_w, l2_2, word_emb, r1, E02);
    spmm128<<<edge_grid(E01), blk, 0, stream>>>(e01_src, e01_dst, e01_w, l1_1, r0s, E01);
    spmm_concat256<<<edge_grid(E02), blk, 0, stream>>>(e02_src, e02_dst, e02_w, l1_2, word_emb, r1s, E02);

    // normalize + concat into outputs
    l2norm_concat<<<dim3((N0 + 7) / 8), blk, 0, stream>>>(r0,  r1,  doc,     N0);
    l2norm_concat<<<dim3((N0 + 7) / 8), blk, 0, stream>>>(r0s, r1s, doc_svd, N0);
}